// CRFOutputLayer_4320737099895
// MI455X (gfx1250) — compile-verified
//
#include <hip/hip_runtime.h>
#include <hip/hip_bf16.h>
#include <stdint.h>

// Problem dims (from reference): B=128, T=512, H=1024, L=32
#define BB 128
#define TT 512
#define HH 1024
#define LL 32
#define MM (BB * TT)          // 65536 GEMM rows

typedef __attribute__((ext_vector_type(2))) float v2f;
typedef __attribute__((ext_vector_type(8))) float v8f;

// ---------------------------------------------------------------------------
// Kernel 1: emissions = outputs @ fc_w^T + fc_b   (M=65536, K=1024, N=32)
// One block = 128 rows x full N=32. 8 waves; each wave owns 16 rows and
// accumulates two 16x16 f32 tiles via V_WMMA_F32_16X16X4_F32.
// Global->LDS staging uses CDNA5 GLOBAL_LOAD_ASYNC_TO_LDS_B128 with software
// double buffering: chunk c+1 is in flight (ASYNCcnt) under chunk c's WMMAs.
// ---------------------------------------------------------------------------
#define TILE_M 128
#define TILE_K 64
#define LDSP   (TILE_K + 4)   // pad 4 floats: conflict-free fragment reads, 16B-aligned rows
#define CHUNKS (HH / TILE_K)  // 16

__global__ __launch_bounds__(256) void emis_gemm_kernel(
    const float* __restrict__ A,    // outputs [MM][HH]
    const float* __restrict__ W,    // fc_w    [LL][HH]
    const float* __restrict__ bias, // fc_b    [LL]
    float* __restrict__ emis)       // out     [MM][LL]
{
    __shared__ __align__(16) float sA[2][TILE_M * LDSP];  // 2 x 34816 B
    __shared__ __align__(16) float sB[2][LL * LDSP];      // 2 x  8704 B

    const int tid  = threadIdx.x;
    const int w    = tid >> 5;        // wave id 0..7
    const int lane = tid & 31;
    const int half = lane >> 4;       // 0: lanes 0-15, 1: lanes 16-31
    const int l16  = lane & 15;
    const int m0   = blockIdx.x * TILE_M;

    v8f c0 = {};
    v8f c1 = {};

    // Issue one K-chunk of async global->LDS copies (10 ASYNCcnt ops per wave:
    // 8 for the 128x64 A slab, 2 for the 32x64 W slab; each lane moves 16 B).
    auto issue_chunk = [&](int k0, int buf) {
        #pragma unroll
        for (int i = 0; i < 8; ++i) {
            const int linear = tid + i * 256;        // b128 unit index
            const int row    = linear >> 4;          // 16 x b128 per row
            const int c4     = (linear & 15) << 2;   // float col
            const uint32_t lds = (uint32_t)(uintptr_t)&sA[buf][row * LDSP + c4];
            const uint32_t go  = (uint32_t)(((uint32_t)(m0 + row) * HH + k0 + c4) * 4u);
            asm volatile("global_load_async_to_lds_b128 %0, %1, %2"
                         :: "v"(lds), "v"(go), "s"(A) : "memory");
        }
        #pragma unroll
        for (int i = 0; i < 2; ++i) {
            const int linear = tid + i * 256;
            const int row    = linear >> 4;
            const int c4     = (linear & 15) << 2;
            const uint32_t lds = (uint32_t)(uintptr_t)&sB[buf][row * LDSP + c4];
            const uint32_t go  = (uint32_t)(((uint32_t)row * HH + k0 + c4) * 4u);
            asm volatile("global_load_async_to_lds_b128 %0, %1, %2"
                         :: "v"(lds), "v"(go), "s"(W) : "memory");
        }
    };

    issue_chunk(0, 0);

    for (int c = 0; c < CHUNKS; ++c) {
        if (c + 1 < CHUNKS) {
            // Prefetch next chunk into the other buffer, then wait until only
            // those 10 newest ops remain (async loads complete in order).
            issue_chunk((c + 1) * TILE_K, (c + 1) & 1);
            asm volatile("s_wait_asynccnt 10" ::: "memory");
        } else {
            asm volatile("s_wait_asynccnt 0" ::: "memory");
        }
        __syncthreads();   // all waves' slabs for chunk c are now in LDS

        const float* cA = &sA[c & 1][0];
        const float* cB = &sB[c & 1][0];
        const int arow = w * 16 + l16;     // this lane's A row (M = l16)
        #pragma unroll
        for (int kk = 0; kk < TILE_K; kk += 4) {
            const int kc = kk + 2 * half;  // K = 2*(lane/16) + v per WMMA layout
            v2f a  = *(const v2f*)(&cA[arow * LDSP + kc]);
            v2f b0 = *(const v2f*)(&cB[l16 * LDSP + kc]);          // N tile 0
            v2f b1 = *(const v2f*)(&cB[(16 + l16) * LDSP + kc]);   // N tile 1
            c0 = __builtin_amdgcn_wmma_f32_16x16x4_f32(
                     false, a, false, b0, (short)0, c0, false, false);
            c1 = __builtin_amdgcn_wmma_f32_16x16x4_f32(
                     false, a, false, b1, (short)0, c1, false, false);
        }
        __syncthreads();   // done reading buf[c&1] before chunk c+2 overwrites it
    }

    // C/D layout: VGPR r -> M = r + 8*half, N = l16 (tile-local). Add bias, store.
    const float bb0 = bias[l16];
    const float bb1 = bias[16 + l16];
    const int mbase = m0 + w * 16 + 8 * half;
    #pragma unroll
    for (int r = 0; r < 8; ++r) {
        emis[(size_t)(mbase + r) * LL + l16]      = c0[r] + bb0;
        emis[(size_t)(mbase + r) * LL + 16 + l16] = c1[r] + bb1;
    }
}

// ---------------------------------------------------------------------------
// Wave reduction helpers (wave32)
// ---------------------------------------------------------------------------
__device__ inline float wave_max(float v) {
    #pragma unroll
    for (int o = 16; o > 0; o >>= 1) v = fmaxf(v, __shfl_xor(v, o, 32));
    return v;
}
__device__ inline float wave_sum(float v) {
    #pragma unroll
    for (int o = 16; o > 0; o >>= 1) v += __shfl_xor(v, o, 32);
    return v;
}
__device__ inline int wave_sum_i(int v) {
    #pragma unroll
    for (int o = 16; o > 0; o >>= 1) v += __shfl_xor(v, o, 32);
    return v;
}

// ---------------------------------------------------------------------------
// Kernel 2: per-sequence CRF. One block (one wave32) per batch element.
// Lane j owns tag state j (L == 32 == wave32, perfect fit).
// ---------------------------------------------------------------------------
__global__ __launch_bounds__(32) void crf_scan_kernel(
    const float* __restrict__ emis,     // [MM][LL]
    const int* __restrict__ labels,     // [BB][TT]
    const uint8_t* __restrict__ mask,   // [BB][TT] (bool)
    const float* __restrict__ startT,   // [LL]
    const float* __restrict__ endT,     // [LL]
    const float* __restrict__ trans,    // [LL][LL]
    float* __restrict__ llh,            // [BB]
    uint8_t* __restrict__ hist,         // [TT-1][BB][LL]
    float* __restrict__ pred)           // [BB][TT] (as float)
{
    __shared__ float sTrans[LL * 33];
    __shared__ float sScore[LL];
    __shared__ float sV[LL];

    const int b    = blockIdx.x;
    const int lane = threadIdx.x;
    const int* lab = labels + (size_t)b * TT;
    const uint8_t* mk = mask + (size_t)b * TT;
    const float* em_b = emis + (size_t)b * TT * LL;

    // Transitions into LDS, padded stride 33 (conflict-free column reads).
    #pragma unroll
    for (int i = 0; i < LL; ++i) sTrans[i * 33 + lane] = trans[i * LL + lane];
    __syncthreads();

    // ---- numerator (gather path), strided over t then wave-reduced ----
    float part = 0.f;
    int   cnt  = 0;
    for (int t = lane; t < TT; t += 32) {
        const int on = (mk[t] != 0);
        cnt += on;
        if (t >= 1 && on) {
            const int lp = lab[t - 1];
            const int lc = lab[t];
            part += sTrans[lp * 33 + lc] + em_b[(size_t)t * LL + lc];
        }
    }
    float num = wave_sum(part);
    const int seq_end = wave_sum_i(cnt) - 1;
    num += startT[lab[0]] + em_b[lab[0]];   // t = 0 emission + start
    num += endT[lab[seq_end]];

    // ---- fused forward-logsumexp + Viterbi scan ----
    float score  = startT[lane] + em_b[lane];   // score0
    float vscore = score;

    for (int t = 1; t < TT; ++t) {
        sScore[lane] = score;
        sV[lane]     = vscore;
        __syncthreads();

        const float em = em_b[(size_t)t * LL + lane];

        // logsumexp over previous states i (two-pass)
        float m = -__builtin_inff();
        #pragma unroll 8
        for (int i = 0; i < LL; ++i)
            m = fmaxf(m, sScore[i] + sTrans[i * 33 + lane]);
        float s = 0.f;
        #pragma unroll 8
        for (int i = 0; i < LL; ++i)
            s += __expf(sScore[i] + sTrans[i * 33 + lane] - m);
        const float nxt = m + __logf(s) + em;

        // Viterbi max/argmax (strict > keeps first index, matches jnp.argmax)
        float bm = -__builtin_inff();
        int   bi = 0;
        #pragma unroll 8
        for (int i = 0; i < LL; ++i) {
            const float v = sV[i] + sTrans[i * 33 + lane];
            if (v > bm) { bm = v; bi = i; }
        }
        const float vnxt = bm + em;

        const bool mt = (mk[t] != 0);
        score  = mt ? nxt  : score;
        vscore = mt ? vnxt : vscore;
        hist[(size_t)(t - 1) * (BB * LL) + (size_t)b * LL + lane] =
            mt ? (uint8_t)bi : (uint8_t)lane;
        __syncthreads();
    }

    // ---- denominator: logsumexp over states of (score + endT) ----
    const float x = score + endT[lane];
    const float mx = wave_max(x);
    const float sx = wave_sum(__expf(x - mx));
    const float denom = mx + __logf(sx);

    if (lane == 0) llh[b] = num - denom;

    // ---- last tag: argmax over states of (vscore + endT) ----
    float bv = vscore + endT[lane];
    int   bidx = lane;
    #pragma unroll
    for (int o = 16; o > 0; o >>= 1) {
        const float ov = __shfl_xor(bv, o, 32);
        const int   oi = __shfl_xor(bidx, o, 32);
        if (ov > bv || (ov == bv && oi < bidx)) { bv = ov; bidx = oi; }
    }

    // ---- backtrace (lane 0, sequential over T) ----
    if (lane == 0) {
        int tag = bidx;
        pred[(size_t)b * TT + (TT - 1)] = (float)tag;
        for (int t = TT - 2; t >= 0; --t) {
            tag = hist[(size_t)t * (BB * LL) + (size_t)b * LL + tag];
            pred[(size_t)b * TT + t] = (float)tag;
        }
    }
}

// ---------------------------------------------------------------------------
// Kernel 3: loss = -mean(llh)
// ---------------------------------------------------------------------------
__global__ __launch_bounds__(128) void loss_reduce_kernel(
    const float* __restrict__ llh, float* __restrict__ out)
{
    __shared__ float tmp[128];
    const int tid = threadIdx.x;
    tmp[tid] = llh[tid];
    __syncthreads();
    #pragma unroll
    for (int s = 64; s > 0; s >>= 1) {
        if (tid < s) tmp[tid] += tmp[tid + s];
        __syncthreads();
    }
    if (tid == 0) out[0] = -tmp[0] / (float)BB;
}

// ---------------------------------------------------------------------------
// Host launch
// ---------------------------------------------------------------------------
extern "C" void kernel_launch(void* const* d_in, const int* in_sizes, int n_in,
                              void* d_out, int out_size, void* d_ws, size_t ws_size,
                              hipStream_t stream) {
    const float*   outputs = (const float*)d_in[0];     // [BB][TT][HH]
    const int*     labels  = (const int*)d_in[1];       // [BB][TT]
    const uint8_t* mask    = (const uint8_t*)d_in[2];   // [BB][TT] bool
    const float*   fc_w    = (const float*)d_in[3];     // [LL][HH]
    const float*   fc_b    = (const float*)d_in[4];     // [LL]
    const float*   startT  = (const float*)d_in[5];     // [LL]
    const float*   endT    = (const float*)d_in[6];     // [LL]
    const float*   trans   = (const float*)d_in[7];     // [LL][LL]

    float* out = (float*)d_out;           // [0] = loss, [1..] = pred (B*T floats)

    // Workspace layout
    char* ws = (char*)d_ws;
    float*   emis = (float*)ws;                                   // 8 MB
    float*   llh  = (float*)(ws + (size_t)MM * LL * sizeof(float));
    uint8_t* hist = (uint8_t*)(ws + (size_t)MM * LL * sizeof(float) + 512);

    emis_gemm_kernel<<<MM / TILE_M, 256, 0, stream>>>(outputs, fc_w, fc_b, emis);

    crf_scan_kernel<<<BB, 32, 0, stream>>>(emis, labels, mask, startT, endT, trans,
                                           llh, hist, out + 1);

    loss_reduce_kernel<<<1, 128, 0, stream>>>(llh, out);
}